// DeformConv_81209241633111
// MI455X (gfx1250) — compile-verified
//
#include <hip/hip_runtime.h>
#include <hip/hip_bf16.h>

typedef __attribute__((ext_vector_type(2))) float v2f;
typedef __attribute__((ext_vector_type(8))) float v8f;

// Problem constants (match reference)
#define DC_KH 3
#define DC_KW 3
#define DC_B  4
#define DC_C  64
#define DC_H  128
#define DC_W  128
#define DC_O  64
#define DC_HO 128
#define DC_WO 128
#define DC_K  (DC_KH * DC_KW)        // 9
#define DC_CK (DC_C * DC_K)          // 576 (GEMM K)
#define DC_NP (DC_HO * DC_WO)        // 16384 pixels per image
#define NTILE 16                     // pixels per block (GEMM N tile)

__global__ __launch_bounds__(128) void deform_conv_wmma_f32(
    const float* __restrict__ x,       // [B, C, H, W]
    const float* __restrict__ offset,  // [B, 2*K, HO, WO]
    const float* __restrict__ weight,  // [O, C, KH, KW] -> [O][c*9+kk]
    const float* __restrict__ bias,    // [O]
    float* __restrict__ out)           // [B, O, HO, WO]
{
    // B-matrix panel in LDS: sampled[n][k], k = c*9+kk.  16 x 576 f32 = 36 KB.
    __shared__ float sB[NTILE][DC_CK];

    const int tid     = threadIdx.x;
    const int b       = blockIdx.y;
    const int pixBase = blockIdx.x * NTILE;

    // ---------------- Stage 1: bilinear sampling into LDS ----------------
    // One (n, kk) pair per thread iteration: compute coords/weights once,
    // then gather+blend for all 64 channels (x is L2-resident: 16.8MB / 192MB L2).
    for (int p = tid; p < NTILE * DC_K; p += 128) {
        const int n   = p / DC_K;
        const int kk  = p % DC_K;
        const int pix = pixBase + n;
        const int ho  = pix / DC_WO;
        const int wo  = pix % DC_WO;

        const size_t obase = (((size_t)b * (2 * DC_K) + 2 * kk) * DC_HO + ho) * DC_WO + wo;
        const float offy = offset[obase];
        const float offx = offset[obase + (size_t)DC_HO * DC_WO];

        // STRIDE=1, PAD=1, DIL=1
        const float py = offy + (float)(kk / DC_KW) + (float)(ho - 1);
        const float px = offx + (float)(kk % DC_KW) + (float)(wo - 1);

        const float y0f = floorf(py), x0f = floorf(px);
        const float ly = py - y0f, lx = px - x0f;
        const float hy = 1.0f - ly, hx = 1.0f - lx;
        const int y0 = (int)y0f, x0 = (int)x0f;
        const int y1 = y0 + 1,  x1 = x0 + 1;

        const float vy0 = (y0 >= 0 && y0 < DC_H) ? 1.0f : 0.0f;
        const float vy1 = (y1 >= 0 && y1 < DC_H) ? 1.0f : 0.0f;
        const float vx0 = (x0 >= 0 && x0 < DC_W) ? 1.0f : 0.0f;
        const float vx1 = (x1 >= 0 && x1 < DC_W) ? 1.0f : 0.0f;

        const float w00 = hy * hx * vy0 * vx0;
        const float w01 = hy * lx * vy0 * vx1;
        const float w10 = ly * hx * vy1 * vx0;
        const float w11 = ly * lx * vy1 * vx1;

        const int iy0 = min(max(y0, 0), DC_H - 1);
        const int iy1 = min(max(y1, 0), DC_H - 1);
        const int ix0 = min(max(x0, 0), DC_W - 1);
        const int ix1 = min(max(x1, 0), DC_W - 1);

        const int i00 = iy0 * DC_W + ix0;
        const int i01 = iy0 * DC_W + ix1;
        const int i10 = iy1 * DC_W + ix0;
        const int i11 = iy1 * DC_W + ix1;

        const float* xb = x + (size_t)b * DC_C * DC_H * DC_W;
        #pragma unroll 4
        for (int c = 0; c < DC_C; ++c) {
            const float* xc = xb + (size_t)c * DC_H * DC_W;
            const float v = w00 * xc[i00] + w01 * xc[i01]
                          + w10 * xc[i10] + w11 * xc[i11];
            sB[n][c * DC_K + kk] = v;
        }
    }
    __syncthreads();

    // ---------------- Stage 2: WMMA FP32 GEMM (16x16x4) ----------------
    // Wave w owns M tile [w*16, w*16+16). D = A(16x4) * B(4x16) + C, K-loop of 144.
    const int wave  = tid >> 5;
    const int lane  = tid & 31;
    const int mBase = wave * 16;
    const int lm    = lane & 15;          // A: row M   | B: col N
    const int khalf = (lane >> 4) << 1;   // lanes 0-15 -> K{0,1}; 16-31 -> K{2,3}

    // A fragment source: weight row, contiguous K (8B-aligned: khalf even).
    const float* wrow = weight + (size_t)(mBase + lm) * DC_CK + khalf;
    __builtin_prefetch(wrow, 0, 0);

    v8f acc = {};
    #pragma unroll 4
    for (int k0 = 0; k0 < DC_CK; k0 += 4) {
        v2f a  = *(const v2f*)(wrow + k0);                 // global_load_b64
        v2f bf = *(const v2f*)(&sB[lm][k0 + khalf]);       // ds_load_b64
        acc = __builtin_amdgcn_wmma_f32_16x16x4_f32(
            /*neg_a=*/false, a, /*neg_b=*/false, bf,
            /*c_mod=*/(short)0, acc, /*reuse_a=*/false, /*reuse_b=*/false);
    }

    // ---------------- Epilogue: D layout -> global, + bias ----------------
    // VGPR r, lanes 0-15: M = r; lanes 16-31: M = r + 8. N = lane & 15.
    const int pix  = pixBase + lm;
    const int mOff = (lane >> 4) << 3;
    #pragma unroll
    for (int r = 0; r < 8; ++r) {
        const int m = mBase + mOff + r;
        out[((size_t)b * DC_O + m) * DC_NP + pix] = acc[r] + bias[m];
    }
}

extern "C" void kernel_launch(void* const* d_in, const int* in_sizes, int n_in,
                              void* d_out, int out_size, void* d_ws, size_t ws_size,
                              hipStream_t stream) {
    const float* x      = (const float*)d_in[0];
    const float* offset = (const float*)d_in[1];
    const float* weight = (const float*)d_in[2];
    const float* bias   = (const float*)d_in[3];
    float* out          = (float*)d_out;

    dim3 grid(DC_NP / NTILE, DC_B);   // (1024, 4)
    deform_conv_wmma_f32<<<grid, 128, 0, stream>>>(x, offset, weight, bias, out);
}